// VectorQuantizer_38397007626791
// MI455X (gfx1250) — compile-verified
//
#include <hip/hip_runtime.h>
#include <hip/hip_bf16.h>

// ---------------- problem constants ----------------
#define BATCH   32
#define CDIM    64
#define HW      1024            // 32*32
#define NROWS   (BATCH * HW)    // 32768
#define KCODES  2048
#define ZELEMS  (NROWS * CDIM)  // 2,097,152
#define OHELEMS ((size_t)BATCH * KCODES * HW) // 67,108,864

// d_out float offsets (tuple concatenated in return order)
#define OFF_LOSS   0
#define OFF_ZQ     1
#define OFF_PERP   (1 + ZELEMS)                      // 2,097,153
#define OFF_ONEHOT (OFF_PERP + 1)                    // 2,097,154
#define OFF_IDX    (OFF_ONEHOT + (size_t)OHELEMS)    // 69,206,018
#define OFF_HIST   (OFF_IDX + NROWS)                 // 69,238,786

typedef float v2f __attribute__((ext_vector_type(2)));
typedef float v8f __attribute__((ext_vector_type(8)));

// ---------------- init: zero accumulators ----------------
__global__ void vq_init(float* hist, float* counts, float* sse) {
    int i = blockIdx.x * 256 + threadIdx.x;
    if (i < BATCH * KCODES) hist[i] = 0.0f;
    if (i < KCODES)         counts[i] = 0.0f;
    if (i == 0)             *sse = 0.0f;
}

// ---------------- transpose z [B,C,HW] -> zf [N,C] ----------------
__global__ void vq_transpose(const float* __restrict__ z, float* __restrict__ zf) {
    size_t e = (size_t)blockIdx.x * 256 + threadIdx.x;   // over z layout
    int hw = (int)(e & (HW - 1));
    int c  = (int)((e >> 10) & (CDIM - 1));
    int b  = (int)(e >> 16);
    zf[((size_t)((b << 10) | hw)) * CDIM + c] = z[e];
}

// ---------------- codebook squared norms ----------------
__global__ void vq_cnorm(const float* __restrict__ cb, float* __restrict__ cnorm) {
    int k = blockIdx.x * 256 + threadIdx.x;
    if (k < KCODES) {
        const float* r = cb + (size_t)k * CDIM;
        float s = 0.0f;
        #pragma unroll 8
        for (int c = 0; c < CDIM; ++c) s += r[c] * r[c];
        cnorm[k] = s;
    }
}

// ---------------- fused distance GEMM + argmin + online softmax ----------------
// One wave = 16 rows. Block = 8 waves = 128 rows (always inside one batch image).
__global__ __launch_bounds__(256) void vq_main(
    const float* __restrict__ zf, const float* __restrict__ cb,
    const float* __restrict__ cnorm,
    int*   __restrict__ idx_out,    // ws: [N] indices
    float* __restrict__ counts,     // ws: [K] code counts
    float* __restrict__ out_idx_f,  // d_out: encoding_indices as float [N]
    float* __restrict__ out_hist)   // d_out: index_histogram [B,K] (zeroed)
{
    __shared__ float lhist[KCODES];
    const int tid  = threadIdx.x;
    const int lane = tid & 31;
    const int wave = tid >> 5;
    const int hi   = lane >> 4;     // C/D layout: rows v (hi=0) / v+8 (hi=1)
    const int lo   = lane & 15;     // code-in-tile for B/C/D; row for A
    const int rowbase = (blockIdx.x * 8 + wave) * 16;
    const int batch   = rowbase >> 10;

    for (int i = tid; i < KCODES; i += 256) lhist[i] = 0.0f;
    __syncthreads();

    // Preload A (16 rows x 64 k): a[s] = zf[rowbase+lo][4s + 2hi + {0,1}]
    v2f a[16];
    const float* zrow = zf + (size_t)(rowbase + lo) * CDIM + 2 * hi;
    float pn = 0.0f;
    #pragma unroll
    for (int s = 0; s < 16; ++s) {
        a[s] = *(const v2f*)(zrow + 4 * s);
        pn += a[s].x * a[s].x + a[s].y * a[s].y;
    }
    float rnorm = pn + __shfl_xor(pn, 16, 32);
    float rn8[8];
    #pragma unroll
    for (int v = 0; v < 8; ++v) rn8[v] = __shfl(rnorm, v + 8 * hi, 32);

    float dmin[8], ssum[8]; int amin[8];
    #pragma unroll
    for (int v = 0; v < 8; ++v) { dmin[v] = 3.0e38f; ssum[v] = 0.0f; amin[v] = 0; }

    // B-tile load: bt[s] = cb[j+lo][4s + 2hi + {0,1}], 16 independent b64 loads
    auto loadB = [&](v2f* bt, int j) {
        const float* crow = cb + (size_t)(j + lo) * CDIM + 2 * hi;
        #pragma unroll
        for (int s = 0; s < 16; ++s) bt[s] = *(const v2f*)(crow + 4 * s);
    };

    v2f b0[16], b1[16];

    // ---- pass 1: WMMA distances + branchless online argmin/logsumexp ----
    auto tile1 = [&](const v2f* bt, int j) {
        const float cnj = cnorm[j + lo];
        v8f acc = {};
        #pragma unroll
        for (int s = 0; s < 16; ++s)
            acc = __builtin_amdgcn_wmma_f32_16x16x4_f32(
                false, a[s], false, bt[s], (short)0, acc, false, false);
        #pragma unroll
        for (int v = 0; v < 8; ++v) {
            float d  = fmaf(-2.0f, acc[v], rn8[v] + cnj);
            bool nmn = d < dmin[v];
            // exactly one of the two logsumexp rescale factors is 1:
            float e  = __expf(-fabsf(d - dmin[v]));
            ssum[v]  = nmn ? fmaf(ssum[v], e, 1.0f) : (ssum[v] + e);
            amin[v]  = nmn ? (j + lo) : amin[v];
            dmin[v]  = fminf(d, dmin[v]);
        }
    };

    loadB(b0, 0);
    for (int j = 0; j < KCODES; j += 32) {
        loadB(b1, j + 16);                       // prefetch next tile (regs)
        tile1(b0, j);
        if (j + 32 < KCODES) loadB(b0, j + 32);  // prefetch tile after next
        tile1(b1, j + 16);
    }

    // ---- cross-lane reduce over the 16 code lanes (logsumexp merge) ----
    #pragma unroll
    for (int v = 0; v < 8; ++v) {
        #pragma unroll
        for (int m = 1; m < 16; m <<= 1) {
            float od = __shfl_xor(dmin[v], m, 32);
            int   oa = __shfl_xor(amin[v], m, 32);
            float os = __shfl_xor(ssum[v], m, 32);
            float nm = fminf(dmin[v], od);
            ssum[v] = ssum[v] * __expf(nm - dmin[v]) + os * __expf(nm - od);
            bool take = (od < dmin[v]) || (od == dmin[v] && oa < amin[v]);
            amin[v] = take ? oa : amin[v];
            dmin[v] = nm;
        }
    }

    if (lo == 0) {
        #pragma unroll
        for (int v = 0; v < 8; ++v) {
            int r = rowbase + v + 8 * hi;
            idx_out[r]   = amin[v];
            out_idx_f[r] = (float)amin[v];
            atomicAdd(&counts[amin[v]], 1.0f);
        }
    }

    float sinv[8];
    #pragma unroll
    for (int v = 0; v < 8; ++v) sinv[v] = 1.0f / ssum[v];

    // ---- pass 2: recompute distances, scatter softmax probs into LDS hist ----
    auto tile2 = [&](const v2f* bt, int j) {
        const float cnj = cnorm[j + lo];
        v8f acc = {};
        #pragma unroll
        for (int s = 0; s < 16; ++s)
            acc = __builtin_amdgcn_wmma_f32_16x16x4_f32(
                false, a[s], false, bt[s], (short)0, acc, false, false);
        float psum = 0.0f;
        #pragma unroll
        for (int v = 0; v < 8; ++v) {
            float d = fmaf(-2.0f, acc[v], rn8[v] + cnj);
            psum += __expf(dmin[v] - d) * sinv[v];
        }
        psum += __shfl_xor(psum, 16, 32);        // lanes l, l+16 share code j+lo
        if (hi == 0) atomicAdd(&lhist[j + lo], psum);   // ds_add_f32
    };

    loadB(b0, 0);
    for (int j = 0; j < KCODES; j += 32) {
        loadB(b1, j + 16);
        tile2(b0, j);
        if (j + 32 < KCODES) loadB(b0, j + 32);
        tile2(b1, j + 16);
    }
    __syncthreads();

    float* gh = out_hist + (size_t)batch * KCODES;
    for (int i = tid; i < KCODES; i += 256) atomicAdd(&gh[i], lhist[i]);
}

// ---------------- gather zq, STE output, MSE accumulation ----------------
__global__ void vq_zq(const float* __restrict__ z, const float* __restrict__ cb,
                      const int* __restrict__ idx, float* __restrict__ out_zq,
                      float* __restrict__ sse) {
    size_t e = (size_t)blockIdx.x * 256 + threadIdx.x;   // [B,C,HW] layout
    int hw = (int)(e & (HW - 1));
    int c  = (int)((e >> 10) & (CDIM - 1));
    int b  = (int)(e >> 16);
    int n  = (b << 10) | hw;
    float zq = cb[(size_t)idx[n] * CDIM + c];
    float zv = z[e];
    float diff = zq - zv;
    out_zq[e] = zv + diff;      // straight-through: z + (zq - z)
    __shared__ float red[256];
    red[threadIdx.x] = diff * diff;
    __syncthreads();
    for (int s = 128; s > 0; s >>= 1) {
        if ((int)threadIdx.x < s) red[threadIdx.x] += red[threadIdx.x + s];
        __syncthreads();
    }
    if (threadIdx.x == 0) atomicAdd(sse, red[0]);
}

// ---------------- one-hot output [B,K,H,W] (256 MB stream) ----------------
__global__ void vq_onehot(const int* __restrict__ idx, float* __restrict__ oh) {
    size_t e = (size_t)blockIdx.x * 256 + threadIdx.x;
    int hw = (int)(e & (HW - 1));
    int k  = (int)((e >> 10) & (KCODES - 1));
    int b  = (int)(e >> 21);
    oh[e] = (idx[(b << 10) | hw] == k) ? 1.0f : 0.0f;
}

// ---------------- loss + perplexity ----------------
__global__ void vq_final(const float* __restrict__ counts, const float* __restrict__ sse,
                         float* __restrict__ out_loss, float* __restrict__ out_perp) {
    __shared__ float red[256];
    float acc = 0.0f;
    for (int k = threadIdx.x; k < KCODES; k += 256) {
        float p = counts[k] * (1.0f / (float)NROWS);
        acc += p * __logf(p + 1e-10f);
    }
    red[threadIdx.x] = acc;
    __syncthreads();
    for (int s = 128; s > 0; s >>= 1) {
        if ((int)threadIdx.x < s) red[threadIdx.x] += red[threadIdx.x + s];
        __syncthreads();
    }
    if (threadIdx.x == 0) {
        *out_perp = __expf(-red[0]);
        // loss = quantization + 0.25*commitment = 1.25 * mean((zq - z)^2)
        *out_loss = 1.25f * (*sse) * (1.0f / (float)ZELEMS);
    }
}

extern "C" void kernel_launch(void* const* d_in, const int* in_sizes, int n_in,
                              void* d_out, int out_size, void* d_ws, size_t ws_size,
                              hipStream_t stream) {
    const float* z  = (const float*)d_in[0];   // [32,64,32,32]
    const float* cb = (const float*)d_in[1];   // [2048,64]
    float* out = (float*)d_out;

    // workspace layout
    float* ws      = (float*)d_ws;
    float* zf      = ws;                                   // N*C floats
    float* cnorm   = ws + ZELEMS;                          // K floats
    int*   idx     = (int*)(ws + ZELEMS + KCODES);         // N ints
    float* counts  = ws + ZELEMS + KCODES + NROWS;         // K floats
    float* sse     = counts + KCODES;                      // 1 float

    float* out_loss = out + OFF_LOSS;
    float* out_zq   = out + OFF_ZQ;
    float* out_perp = out + OFF_PERP;
    float* out_oh   = out + OFF_ONEHOT;
    float* out_idx  = out + OFF_IDX;
    float* out_hist = out + OFF_HIST;

    vq_init<<<(BATCH * KCODES + 255) / 256, 256, 0, stream>>>(out_hist, counts, sse);
    vq_transpose<<<ZELEMS / 256, 256, 0, stream>>>(z, zf);
    vq_cnorm<<<(KCODES + 255) / 256, 256, 0, stream>>>(cb, cnorm);
    vq_main<<<NROWS / 128, 256, 0, stream>>>(zf, cb, cnorm, idx, counts, out_idx, out_hist);
    vq_zq<<<ZELEMS / 256, 256, 0, stream>>>(z, cb, idx, out_zq, sse);
    vq_onehot<<<(int)(OHELEMS / 256), 256, 0, stream>>>(idx, out_oh);
    vq_final<<<1, 256, 0, stream>>>(counts, sse, out_loss, out_perp);
}